// MarkovChain_81655918231782
// MI455X (gfx1250) — compile-verified
//
#include <hip/hip_runtime.h>
#include <stdint.h>

// ---- CDNA5 vector types ----
typedef __attribute__((ext_vector_type(2))) float        v2f;
typedef __attribute__((ext_vector_type(8))) float        v8f;
typedef __attribute__((ext_vector_type(4))) unsigned int u32x4;
typedef __attribute__((ext_vector_type(8))) int          i32x8;
typedef __attribute__((ext_vector_type(4))) int          i32x4;

#define NSTATES 8192
#define NSTEPS  512
#define BATCH   1024
#define TPB     256   // 8 waves (wave32)

// ============================================================================
// Kernel A: per-row logsumexp of T (rows 0..8191) and of init vector (row 8192)
// One block per row. Row is bulk-staged to LDS via the Tensor Data Mover,
// then each thread register-caches 32 floats (8 x ds b128). Block max is
// broadcast so all 256 partial exp-sums share one max -> combine by plain
// addition, which we route through V_WMMA_F32_16X16X4_F32 (ones-matrix B).
// ============================================================================
__global__ __launch_bounds__(TPB) void lse_rows(const float* __restrict__ T,
                                                const float* __restrict__ initv,
                                                float* __restrict__ lse) {
  __shared__ float rowbuf[NSTATES];   // 32 KB staged row
  __shared__ float red[TPB];          // reduction scratch

  const int tid = threadIdx.x;
  const int row = blockIdx.x;
  const float* rowp = (row < NSTATES) ? (T + (size_t)row * NSTATES) : initv;

#if __has_builtin(__builtin_amdgcn_tensor_load_to_lds)
  // ---- TDM: one descriptor copies the whole 32KB row global -> LDS ----
  if (tid == 0) {
    const uint64_t ga = (uint64_t)(uintptr_t)rowp;
    const uint32_t la = (uint32_t)(uintptr_t)&rowbuf[0];
    // D# group0: count=1 | lds_addr | global_addr[56:0] | type=2 (bits 127:126)
    u32x4 g0 = { 1u, la, (uint32_t)ga, (uint32_t)(ga >> 32) | (2u << 30) };
    // D# group1: data_size=4B (code 2); tensor_dim0=8192; tensor_dim1=1;
    //            tile_dim0=8192; tile_dim1=1; tensor_dim0_stride=8192
    i32x8 g1 = { 0x20000,                        // [17:16] data_size=2
                 (int)((unsigned)NSTATES << 16), // tensor_dim0 lo16 @ [31:16]
                 (1 << 16),                      // tensor_dim1 lo16 = 1
                 (int)((unsigned)NSTATES << 16), // tile_dim0 @ [31:16]
                 1,                              // tile_dim1 = 1
                 NSTATES,                        // tensor_dim0_stride lo32
                 0, 0 };
    i32x4 gz4 = { 0, 0, 0, 0 };
    i32x8 gz8 = { 0, 0, 0, 0, 0, 0, 0, 0 };
    __builtin_amdgcn_tensor_load_to_lds(g0, g1, gz4, gz4, gz8, 0);
#if __has_builtin(__builtin_amdgcn_s_wait_tensorcnt)
    __builtin_amdgcn_s_wait_tensorcnt(0);
#else
    asm volatile("s_wait_tensorcnt 0x0" ::: "memory");
#endif
  }
  __syncthreads();
#else
  // Fallback staging: coalesced b128 copies
  {
    float4* rb4 = (float4*)rowbuf;
    const float4* g4 = (const float4*)rowp;
#pragma unroll
    for (int i = 0; i < 8; ++i) rb4[i * TPB + tid] = g4[i * TPB + tid];
  }
  __syncthreads();
#endif

  // ---- register-cache 32 floats from LDS (8 x ds_load_b128) ----
  float4 v[8];
  {
    const float4* rb4 = (const float4*)rowbuf;
#pragma unroll
    for (int i = 0; i < 8; ++i) v[i] = rb4[i * TPB + tid];
  }

  // ---- block max ----
  float m = v[0].x;
#pragma unroll
  for (int i = 0; i < 8; ++i)
    m = fmaxf(m, fmaxf(fmaxf(v[i].x, v[i].y), fmaxf(v[i].z, v[i].w)));
  red[tid] = m;
  __syncthreads();
  for (int s = TPB / 2; s > 0; s >>= 1) {
    if (tid < s) red[tid] = fmaxf(red[tid], red[tid + s]);
    __syncthreads();
  }
  const float M = red[0];
  __syncthreads();

  // ---- per-thread sum of exp(v - M); common M => partials add linearly ----
  float ssum = 0.0f;
#pragma unroll
  for (int i = 0; i < 8; ++i) {
    ssum += __expf(v[i].x - M) + __expf(v[i].y - M) +
            __expf(v[i].z - M) + __expf(v[i].w - M);
  }
  red[tid] = ssum;
  __syncthreads();
  // fold 256 -> 64
  if (tid < 64) red[tid] = red[tid] + red[tid + 64] + red[tid + 128] + red[tid + 192];
  __syncthreads();

  // ---- 64 -> 1 via V_WMMA_F32_16X16X4_F32 with B = ones (wave 0 only) ----
  if (tid < 32) {
    // A layout (16x4 f32): lanes 0-15 hold K=0,1 for row M=lane in vgpr0/1;
    // lanes 16-31 hold K=2,3 for row M=lane-16. Map A[m][k] = red[4m+k].
    const int mrow = tid & 15;
    const int k0 = (tid >> 4) << 1;
    v2f a; a.x = red[4 * mrow + k0]; a.y = red[4 * mrow + k0 + 1];
    v2f b; b.x = 1.0f; b.y = 1.0f;       // all-ones B => D[m][n] = rowsum(m)
    v8f c = {};
    v8f d = __builtin_amdgcn_wmma_f32_16x16x4_f32(
        /*neg_a=*/false, a, /*neg_b=*/false, b,
        /*c_mod=*/(short)0, c, /*reuse_a=*/false, /*reuse_b=*/false);
    // lanes 0-15: vgpr r = rowsum(r); lanes 16-31: vgpr r = rowsum(r+8)
    float t = d[0] + d[1] + d[2] + d[3] + d[4] + d[5] + d[6] + d[7];
    t += __shfl_xor(t, 16, 32);          // add rows 8..15 half
    if (tid == 0) lse[row] = M + logf(t);
  }
}

// ============================================================================
// Kernel B: one wave32 per batch row; lanes stride the 511 transitions,
// gather T[i,j] - lse[i], wave-reduce, add initial term.
// ============================================================================
__global__ __launch_bounds__(TPB) void chain_ll(const int* __restrict__ x,
                                                const float* __restrict__ T,
                                                const float* __restrict__ initv,
                                                const float* __restrict__ lse,
                                                float* __restrict__ out) {
  const int tid  = threadIdx.x;
  const int lane = tid & 31;
  const int wave = tid >> 5;
  const int b = blockIdx.x * (TPB / 32) + wave;
  if (b >= BATCH) return;
  const int* xb = x + (size_t)b * NSTEPS;

  float acc = 0.0f;
  for (int t = lane; t < NSTEPS - 1; t += 32) {
    const int i = xb[t];
    const int j = xb[t + 1];
    acc += T[(size_t)i * NSTATES + j] - lse[i];
  }
#pragma unroll
  for (int off = 16; off > 0; off >>= 1) acc += __shfl_xor(acc, off, 32);
  if (lane == 0) out[b] = acc + initv[xb[0]] - lse[NSTATES];
}

// ============================================================================
extern "C" void kernel_launch(void* const* d_in, const int* in_sizes, int n_in,
                              void* d_out, int out_size, void* d_ws, size_t ws_size,
                              hipStream_t stream) {
  const int*   x     = (const int*)d_in[0];     // (1024, 512) int32
  const float* initv = (const float*)d_in[1];   // (8192,) f32
  const float* T     = (const float*)d_in[2];   // (8192, 8192) f32
  float* lse = (float*)d_ws;                    // 8193 floats scratch
  float* out = (float*)d_out;                   // (1024,) f32

  lse_rows<<<NSTATES + 1, TPB, 0, stream>>>(T, initv, lse);
  chain_ll<<<BATCH / (TPB / 32), TPB, 0, stream>>>(x, T, initv, lse, out);
}